// BatchSplitFF_32418413150503
// MI455X (gfx1250) — compile-verified
//
#include <hip/hip_runtime.h>

// ---- problem constants (fixed by reference) ----
#define DM     1024   // model dim d
#define NEXP   64     // experts e
#define ESZ    64     // expert hidden f
#define TSLOT  16     // sparsity slots t
#define NGRP   256    // groups b = 4096/16
#define NTOK   4096   // tokens per batch
#define NBATCH 2

typedef __attribute__((ext_vector_type(2))) float v2f;
typedef __attribute__((ext_vector_type(8))) float v8f;

// WMMA f32 16x16x4 helper: D = A(16x4) * B(4x16) + C
__device__ __forceinline__ v8f wmma_f32_4(v2f a, v2f b, v8f c) {
  return __builtin_amdgcn_wmma_f32_16x16x4_f32(false, a, false, b, (short)0, c,
                                               false, false);
}

// CDNA5 async direct-to-LDS copy (16B), tracked by ASYNCcnt.
// LDS target given as wave-relative byte offset (generic ptr low 32 bits).
__device__ __forceinline__ void async_cp_b128(unsigned lds_off,
                                              const float* gptr) {
  asm volatile("global_load_async_to_lds_b128 %0, %1, off"
               :
               : "v"(lds_off), "v"(gptr)
               : "memory");
}
#define WAIT_ASYNC(n) asm volatile("s_wait_asynccnt %0" ::"n"(n) : "memory")

__device__ __forceinline__ unsigned lds_off_of(const void* p) {
  return (unsigned)(uintptr_t)p;  // flat LDS aperture: addr[31:0] = LDS offset
}

// ------------------------------------------------------------------
// Kernel 0: zero the scatter target
// ------------------------------------------------------------------
__global__ void bsff_zero(float4* __restrict__ out, int n4) {
  int i = blockIdx.x * blockDim.x + threadIdx.x;
  if (i < n4) out[i] = make_float4(0.f, 0.f, 0.f, 0.f);
}

// ------------------------------------------------------------------
// Kernel 1: logits[batch][t][e][b] = x_row . controller[:,e] + cbias[e]
// GEMM [8192 x 1024] @ [1024 x 64] via v_wmma_f32_16x16x4_f32.
// A staged in double-buffered LDS chunks via async-to-LDS copies.
// ------------------------------------------------------------------
__global__ __launch_bounds__(128) void bsff_logits(
    const float* __restrict__ x, const float* __restrict__ ctrl,
    const float* __restrict__ cbias, float* __restrict__ lgts) {
  __shared__ __align__(16) float lA[2][16 * 256];

  const int tid  = threadIdx.x;
  const int lane = tid & 31;
  const int wave = tid >> 5;

  const int rowBase = blockIdx.x << 4;          // global row in [0, 8192)
  const int batch   = rowBase >> 12;
  const int b       = (rowBase & (NTOK - 1)) >> 4;
  const float* xrows = x + (size_t)rowBase * DM;

  const int m  = lane & 15;                     // A-frag row
  const int kk = (lane >> 4) << 1;              // A/B-frag K sub-offset
  const int lo = lane & 15;
  const int n  = (wave << 4) + lo;              // expert column

  const unsigned base0 = lds_off_of(&lA[0][0]);
  const unsigned base1 = lds_off_of(&lA[1][0]);

  // issue one 16x256-float chunk (8 async b128 per thread, per wave: 8 ops)
  auto issue = [&](int k0, unsigned base) {
    for (int i = tid; i < 16 * 64; i += 128) {
      const int r = i >> 6, c = i & 63;
      async_cp_b128(base + (unsigned)(((r << 6) + c) << 4),
                    xrows + (size_t)r * DM + k0 + (c << 2));
    }
  };

  v8f acc = {0.f, 0.f, 0.f, 0.f, 0.f, 0.f, 0.f, 0.f};

  issue(0, base0);
  for (int c = 0; c < 4; ++c) {
    const int k0 = c << 8;
    if (c < 3) {
      issue(k0 + 256, (c & 1) ? base0 : base1);
      WAIT_ASYNC(8);                            // chunk c retired (in-order)
    } else {
      WAIT_ASYNC(0);
    }
    __syncthreads();                            // chunk c visible to all waves
    const float* As = &lA[c & 1][0];
    __builtin_prefetch(ctrl + (size_t)(k0) * NEXP, 0, 0);
#pragma unroll 4
    for (int k = 0; k < 256; k += 4) {
      v2f a, bf;
      a.x = As[(m << 8) + k + kk];
      a.y = As[(m << 8) + k + kk + 1];
      const float* bp = ctrl + (size_t)(k0 + k + kk) * NEXP + n;
      bf.x = bp[0];
      bf.y = bp[NEXP];
      acc = wmma_f32_4(a, bf, acc);
    }
    __syncthreads();                            // release buffer c&1
  }

  const float cb = cbias[n];
#pragma unroll
  for (int r = 0; r < 8; ++r) {
    const int t = r + ((lane >> 4) << 3);       // C layout: M = r + 8*(lane>=16)
    lgts[(((size_t)batch * TSLOT + t) * NEXP + n) * NGRP + b] = acc[r] + cb;
  }
}

// ------------------------------------------------------------------
// Kernel 2: winner[batch][t][e] = argmax_b logits[batch][t][e][b]
// (tie term in reference is constant over b -> irrelevant to argmax)
// ------------------------------------------------------------------
__global__ void bsff_argmax(const float* __restrict__ lgts,
                            int* __restrict__ win) {
  int idx = blockIdx.x * blockDim.x + threadIdx.x;  // [0, 2048)
  if (idx >= NBATCH * TSLOT * NEXP) return;
  const float* p = lgts + (size_t)idx * NGRP;
  float best = p[0];
  int bi = 0;
  for (int b = 1; b < NGRP; ++b) {
    float v = p[b];
    if (v > best) { best = v; bi = b; }
  }
  win[idx] = bi;
}

// ------------------------------------------------------------------
// Kernel 3: per (batch, expert) WG:
//   async-gather 16 winning token rows -> LDS (double-buffered K chunks)
//   G = A(16x1024) @ f1[:,e,:]          (WMMA)
//   P = relu(merge-by-winner(G) + bias) (duplicate winners share pre-act sum)
//   O = P(16x64) @ f2[e]                (WMMA)
//   atomic-scatter O rows to out[winner*16 + t]
// ------------------------------------------------------------------
__global__ __launch_bounds__(128) void bsff_expert(
    const float* __restrict__ x, const float* __restrict__ f1,
    const float* __restrict__ bias, const float* __restrict__ f2,
    const int* __restrict__ win, float* __restrict__ out) {
  __shared__ __align__(16) float lA[2][16 * 256];
  __shared__ float lG[16 * 64];
  __shared__ float lP[16 * 64];
  __shared__ int lWin[16];

  const int tid  = threadIdx.x;
  const int lane = tid & 31;
  const int wave = tid >> 5;
  const int e     = blockIdx.x & 63;
  const int batch = blockIdx.x >> 6;

  if (tid < 16) lWin[tid] = win[batch * (TSLOT * NEXP) + tid * NEXP + e];
  __syncthreads();

  const int m  = lane & 15;
  const int kk = (lane >> 4) << 1;
  const int lo = lane & 15;
  const int n1 = (wave << 4) + lo;              // f column, 0..63

  const float* xb = x + (size_t)batch * NTOK * DM;
  const unsigned base0 = lds_off_of(&lA[0][0]);
  const unsigned base1 = lds_off_of(&lA[1][0]);

  // async-gather one K-chunk of the 16 winner rows
  auto issue = [&](int k0, unsigned base) {
    for (int i = tid; i < 16 * 64; i += 128) {
      const int r = i >> 6, c = i & 63;
      const int srcRow = lWin[r] * 16 + r;      // gathered token row
      async_cp_b128(base + (unsigned)(((r << 6) + c) << 4),
                    xb + (size_t)srcRow * DM + k0 + (c << 2));
    }
  };

  // ---- layer 1: G[16x64] = A[16x1024] @ f1[:,e,:] ----
  v8f acc = {0.f, 0.f, 0.f, 0.f, 0.f, 0.f, 0.f, 0.f};
  issue(0, base0);
  for (int c = 0; c < 4; ++c) {
    const int k0 = c << 8;
    if (c < 3) {
      issue(k0 + 256, (c & 1) ? base0 : base1);
      WAIT_ASYNC(8);
    } else {
      WAIT_ASYNC(0);
    }
    __syncthreads();
    const float* As = &lA[c & 1][0];
    __builtin_prefetch(f1 + (size_t)k0 * (NEXP * ESZ) + e * ESZ, 0, 0);
#pragma unroll 4
    for (int k = 0; k < 256; k += 4) {
      v2f a, bf;
      a.x = As[(m << 8) + k + kk];
      a.y = As[(m << 8) + k + kk + 1];
      const float* bp = f1 + (size_t)(k0 + k + kk) * (NEXP * ESZ) + e * ESZ + n1;
      bf.x = bp[0];
      bf.y = bp[NEXP * ESZ];
      acc = wmma_f32_4(a, bf, acc);
    }
    __syncthreads();
  }
#pragma unroll
  for (int r = 0; r < 8; ++r) {
    const int t = r + ((lane >> 4) << 3);
    lG[t * 64 + n1] = acc[r];
  }
  __syncthreads();

  // ---- merge duplicate winners + bias + relu ----
  for (int i = tid; i < 16 * 64; i += 128) {
    const int t = i >> 6, f = i & 63;
    const int wt = lWin[t];
    float s = bias[e * ESZ + f];
#pragma unroll
    for (int t2 = 0; t2 < 16; ++t2)
      if (lWin[t2] == wt) s += lG[t2 * 64 + f];
    lP[i] = fmaxf(s, 0.f);
  }
  __syncthreads();

  // ---- layer 2: O[16x1024] = P @ f2[e] ----
  const float* f2e = f2 + (size_t)e * ESZ * DM;
  for (int tile = wave; tile < 64; tile += 4) {
    const int n0 = tile << 4;
    v8f c2 = {0.f, 0.f, 0.f, 0.f, 0.f, 0.f, 0.f, 0.f};
#pragma unroll
    for (int k = 0; k < ESZ; k += 4) {
      v2f a, bf;
      a.x = lP[(m << 6) + k + kk];
      a.y = lP[(m << 6) + k + kk + 1];
      const float* bp = f2e + (size_t)(k + kk) * DM + n0 + lo;
      bf.x = bp[0];
      bf.y = bp[DM];
      c2 = wmma_f32_4(a, bf, c2);
    }
#pragma unroll
    for (int r = 0; r < 8; ++r) {
      const int t = r + ((lane >> 4) << 3);
      const size_t orow = (size_t)batch * NTOK + (size_t)lWin[t] * 16 + t;
      atomicAdd(out + orow * DM + n0 + lo, c2[r]);
    }
  }
}

// ------------------------------------------------------------------
extern "C" void kernel_launch(void* const* d_in, const int* in_sizes, int n_in,
                              void* d_out, int out_size, void* d_ws,
                              size_t ws_size, hipStream_t stream) {
  const float* x     = (const float*)d_in[0];
  const float* ctrl  = (const float*)d_in[1];
  const float* cbias = (const float*)d_in[2];
  const float* f1    = (const float*)d_in[3];
  const float* bias  = (const float*)d_in[4];
  const float* f2    = (const float*)d_in[5];
  // d_in[6] = sparsity scalar (fixed 16) -- compile-time constant here
  float* out = (float*)d_out;

  // workspace layout: logits [2][16][64][256] f32 (2 MB), then winners (8 KB)
  float* lgts = (float*)d_ws;
  int* win = (int*)((char*)d_ws +
                    (size_t)NBATCH * TSLOT * NEXP * NGRP * sizeof(float));

  const int n4 = NBATCH * NTOK * DM / 4;
  bsff_zero<<<(n4 + 255) / 256, 256, 0, stream>>>((float4*)out, n4);
  bsff_logits<<<(NBATCH * NTOK) / 16, 128, 0, stream>>>(x, ctrl, cbias, lgts);
  bsff_argmax<<<(NBATCH * TSLOT * NEXP + 255) / 256, 256, 0, stream>>>(lgts,
                                                                       win);
  bsff_expert<<<NBATCH * NEXP, 128, 0, stream>>>(x, f1, bias, f2, win, out);
}